// ADVI_23167053595356
// MI455X (gfx1250) — compile-verified
//
#include <hip/hip_runtime.h>
#include <hip/hip_bf16.h>
#include <math.h>

#define N_K   256
#define N_T   128
#define N_C   32
#define N_D   10
#define N_SPK 1000000

#define LOG2PI_F 1.8378770664093453f

typedef __attribute__((ext_vector_type(2))) float v2f;
typedef __attribute__((ext_vector_type(8))) float v8f;

#define KPAD    68           // 66 features padded to multiple of 4
#define KCHUNKS (KPAD / 4)   // 17 WMMA K-steps of 4

// workspace layout (in floats)
#define WS_W_OFF    0                        // 68*32 = 2176 floats (padded region)
#define WS_LP_OFF   8192                     // 32768*32 = 1,048,576 floats
#define WS_PART_OFF (8192 + N_K * N_T * N_C) // block partials + prior slot

#define WAVES_PER_BLOCK  4
#define SPIKES_PER_WAVE  32
#define SPIKES_PER_BLOCK (WAVES_PER_BLOCK * SPIKES_PER_WAVE) // 128
#define MAIN_BLOCKS ((N_SPK + SPIKES_PER_BLOCK - 1) / SPIKES_PER_BLOCK) // 7813
#define NPART (MAIN_BLOCKS + 1)

// ---------------------------------------------------------------------------
// Kernel 0: per component c, Cholesky -> P = cov^{-1}, logdet, build W[68][32]
// ll[n,c] = f_n . W_c with features f = [s_i^2 (10), s_i*s_j i<j (45), s_i (10), 1, 0, 0]
// ---------------------------------------------------------------------------
__global__ void k_build_w(const float* __restrict__ means,
                          const float* __restrict__ covs,
                          float* __restrict__ W) {
    int c = threadIdx.x;
    if (c >= N_C) return;

    float L[N_D][N_D];
    for (int i = 0; i < N_D; i++)
        for (int j = 0; j < N_D; j++)
            L[i][j] = covs[c * N_D * N_D + i * N_D + j];

    // in-place lower Cholesky
    for (int i = 0; i < N_D; i++) {
        for (int j = 0; j <= i; j++) {
            float acc = L[i][j];
            for (int k = 0; k < j; k++) acc -= L[i][k] * L[j][k];
            if (i == j) L[i][i] = sqrtf(acc);
            else        L[i][j] = acc / L[j][j];
        }
    }
    float logdet = 0.f;
    for (int i = 0; i < N_D; i++) logdet += 2.f * logf(L[i][i]);

    // Li = inv(L) (lower triangular, forward substitution)
    float Li[N_D][N_D];
    for (int col = 0; col < N_D; col++) {
        for (int i = 0; i < N_D; i++) {
            if (i < col) { Li[i][col] = 0.f; continue; }
            float acc = (i == col) ? 1.f : 0.f;
            for (int k = col; k < i; k++) acc -= L[i][k] * Li[k][col];
            Li[i][col] = acc / L[i][i];
        }
    }
    // P = Li^T * Li
    float P[N_D][N_D];
    for (int i = 0; i < N_D; i++)
        for (int j = 0; j < N_D; j++) {
            float acc = 0.f;
            for (int k = 0; k < N_D; k++) acc += Li[k][i] * Li[k][j];
            P[i][j] = acc;
        }

    float mu[N_D], Pmu[N_D];
    for (int i = 0; i < N_D; i++) mu[i] = means[c * N_D + i];
    float muPmu = 0.f;
    for (int i = 0; i < N_D; i++) {
        float acc = 0.f;
        for (int j = 0; j < N_D; j++) acc += P[i][j] * mu[j];
        Pmu[i] = acc;
        muPmu += mu[i] * acc;
    }
    float cnst = -0.5f * ((float)N_D * LOG2PI_F + logdet) - 0.5f * muPmu;

    int idx = 0;
    for (int i = 0; i < N_D; i++) W[(idx++) * N_C + c] = -0.5f * P[i][i];
    for (int i = 0; i < N_D; i++)
        for (int j = i + 1; j < N_D; j++) W[(idx++) * N_C + c] = -P[i][j];
    for (int i = 0; i < N_D; i++) W[(idx++) * N_C + c] = Pmu[i];
    W[(idx++) * N_C + c] = cnst;     // feature 65 == 1
    W[66 * N_C + c] = 0.f;           // K padding
    W[67 * N_C + c] = 0.f;
}

// ---------------------------------------------------------------------------
// Kernel 1: log_pis[kt][c] = log_softmax_c( b[c] + beta[c,t] * y[k,t] )
// ---------------------------------------------------------------------------
__global__ void k_logpis(const float* __restrict__ y,
                         const float* __restrict__ b_mu,
                         const float* __restrict__ beta_mu,
                         float* __restrict__ lp) {
    int kt = blockIdx.x * blockDim.x + threadIdx.x;
    if (kt >= N_K * N_T) return;
    int k = kt >> 7;
    int t = kt & (N_T - 1);
    float yv = y[k * N_T + t];

    float lam[N_C];
    float m = -3.4e38f;
#pragma unroll
    for (int c = 0; c < N_C; c++) {
        lam[c] = b_mu[c] + beta_mu[c * N_T + t] * yv;
        m = fmaxf(m, lam[c]);
    }
    float z = 0.f;
#pragma unroll
    for (int c = 0; c < N_C; c++) z += __expf(lam[c] - m);
    float lse = m + __logf(z);
#pragma unroll
    for (int c = 0; c < N_C; c++) lp[kt * N_C + c] = lam[c] - lse;
}

// ---------------------------------------------------------------------------
// Kernel 2: log_prior - log_q  (at the variational mean: z = 0)
//   term per element = -0.5*mu^2 + log_sig
// ---------------------------------------------------------------------------
__global__ void k_prior(const float* __restrict__ b_mu,
                        const float* __restrict__ b_log_sig,
                        const float* __restrict__ beta_mu,
                        const float* __restrict__ beta_log_sig,
                        float* __restrict__ part) {
    int tid = threadIdx.x;
    float acc = 0.f;
    const int total = N_C + N_C * N_T;
    for (int i = tid; i < total; i += blockDim.x) {
        float muv, lsv;
        if (i < N_C) { muv = b_mu[i];        lsv = b_log_sig[i]; }
        else         { muv = beta_mu[i-N_C]; lsv = beta_log_sig[i-N_C]; }
        acc += -0.5f * muv * muv + lsv;
    }
    __shared__ float red[256];
    red[tid] = acc;
    __syncthreads();
    for (int s = 128; s > 0; s >>= 1) {
        if (tid < s) red[tid] += red[tid + s];
        __syncthreads();
    }
    if (tid == 0) part[MAIN_BLOCKS] = red[0];
}

// ---------------------------------------------------------------------------
// Kernel 3 (main): per wave, 32 spikes. Features -> LDS -> WMMA f32 16x16x4
// GEMM (16 spikes x 32 comps per pair of N-tiles), then gather log_pis,
// softmax over c via 16-lane shuffles, accumulate r*(logits - safe_log(r)).
// ---------------------------------------------------------------------------
__launch_bounds__(WAVES_PER_BLOCK * 32)
__global__ void k_main(const float* __restrict__ s,
                       const int*   __restrict__ ks,
                       const int*   __restrict__ ts,
                       const float* __restrict__ W,
                       const float* __restrict__ lp,
                       float* __restrict__ part) {
    __shared__ float feat[WAVES_PER_BLOCK][SPIKES_PER_WAVE][70]; // stride 70: conflict-free
    __shared__ float red[WAVES_PER_BLOCK * 32];

    const int tid  = threadIdx.x;
    const int wv   = tid >> 5;
    const int lane = tid & 31;
    const int half = lane >> 4;   // 0: lanes 0-15, 1: lanes 16-31
    const int lc   = lane & 15;

    // B fragments of W, kept in VGPRs for the whole kernel.
    // B 4x16 f32 layout: v0 = row K=0 (lanes 0-15) / K=2 (lanes 16-31),
    //                    v1 = row K=1 (lanes 0-15) / K=3 (lanes 16-31).
    v2f bfrag[2][KCHUNKS];
#pragma unroll
    for (int nt = 0; nt < 2; nt++) {
#pragma unroll
        for (int kc = 0; kc < KCHUNKS; kc++) {
            int kk = 4 * kc + 2 * half;
            int cc = nt * 16 + lc;
            v2f b;
            b.x = W[kk * N_C + cc];
            b.y = W[(kk + 1) * N_C + cc];
            bfrag[nt][kc] = b;
        }
    }

    float acc = 0.f;
    long group     = (long)blockIdx.x * WAVES_PER_BLOCK + wv;
    long spikeBase = group * SPIKES_PER_WAVE;

    if (spikeBase < N_SPK) {     // N_SPK % 32 == 0: active waves are always full
        int n = (int)spikeBase + lane;

        // my spike's row in the log_pis table
        int myRow = (ks[n] * N_T + ts[n]) * N_C;

        // load s[n][0..9] (row is 8B-aligned: n*40 bytes)
        const float* sp = s + (long)n * N_D;
        float sv[N_D];
#pragma unroll
        for (int i = 0; i < N_D; i += 2) {
            v2f v = *(const v2f*)(sp + i);
            sv[i] = v.x; sv[i + 1] = v.y;
        }

        // build 66 features into LDS (A-matrix staging)
        float* frow = &feat[wv][lane][0];
        int idx = 0;
#pragma unroll
        for (int i = 0; i < N_D; i++) frow[idx++] = sv[i] * sv[i];
#pragma unroll
        for (int i = 0; i < N_D; i++)
#pragma unroll
            for (int j = i + 1; j < N_D; j++) frow[idx++] = sv[i] * sv[j];
#pragma unroll
        for (int i = 0; i < N_D; i++) frow[idx++] = sv[i];
        frow[65] = 1.f; frow[66] = 0.f; frow[67] = 0.f;

        // in-wave LDS handoff: all lanes' DS writes complete before reads
        asm volatile("s_wait_dscnt 0x0" ::: "memory");

#pragma unroll
        for (int mt = 0; mt < 2; mt++) {          // two M-tiles of 16 spikes
            v8f c0 = {};                          // comps 0..15
            v8f c1 = {};                          // comps 16..31
#pragma unroll
            for (int kc = 0; kc < KCHUNKS; kc++) {
                // A 16x4 f32 layout: lanes 0-15 hold K=0(v0),1(v1); lanes 16-31 K=2,3
                int mrow = mt * 16 + lc;
                int kk   = 4 * kc + 2 * half;
                v2f a = *(const v2f*)&feat[wv][mrow][kk];
                c0 = __builtin_amdgcn_wmma_f32_16x16x4_f32(
                        false, a, false, bfrag[0][kc], (short)0, c0, false, false);
                c1 = __builtin_amdgcn_wmma_f32_16x16x4_f32(
                        false, a, false, bfrag[1][kc], (short)0, c1, false, false);
            }

            // C/D layout: VGPR r -> M=r (lanes 0-15), M=r+8 (lanes 16-31); N = lc
#pragma unroll
            for (int r = 0; r < 8; r++) {
                int localSpike = mt * 16 + r + 8 * half;
                int row = __shfl(myRow, localSpike, 32);
                float lg0 = c0[r] + lp[row + lc];
                float lg1 = c1[r] + lp[row + 16 + lc];

                // softmax over 32 comps spread across 16 lanes x 2 regs
                float mx = fmaxf(lg0, lg1);
#pragma unroll
                for (int off = 8; off > 0; off >>= 1)
                    mx = fmaxf(mx, __shfl_xor(mx, off, 16));
                float e0 = __expf(lg0 - mx);
                float e1 = __expf(lg1 - mx);
                float zs = e0 + e1;
#pragma unroll
                for (int off = 8; off > 0; off >>= 1)
                    zs += __shfl_xor(zs, off, 16);
                float inv = 1.0f / zs;
                float r0 = e0 * inv;
                float r1 = e1 * inv;
                acc += r0 * (lg0 - __logf(r0 + 1e-10f))
                     + r1 * (lg1 - __logf(r1 + 1e-10f));
            }
        }
    }

    // deterministic block reduction
    red[tid] = acc;
    __syncthreads();
    for (int st = (WAVES_PER_BLOCK * 32) / 2; st > 0; st >>= 1) {
        if (tid < st) red[tid] += red[tid + st];
        __syncthreads();
    }
    if (tid == 0) part[blockIdx.x] = red[0];
}

// ---------------------------------------------------------------------------
// Kernel 4: deterministic final reduction of block partials (+ prior slot)
// ---------------------------------------------------------------------------
__global__ void k_final(const float* __restrict__ part, float* __restrict__ out) {
    int tid = threadIdx.x;
    float acc = 0.f;
    for (int i = tid; i < NPART; i += 256) acc += part[i];
    __shared__ float red[256];
    red[tid] = acc;
    __syncthreads();
    for (int s = 128; s > 0; s >>= 1) {
        if (tid < s) red[tid] += red[tid + s];
        __syncthreads();
    }
    if (tid == 0) out[0] = red[0];
}

extern "C" void kernel_launch(void* const* d_in, const int* in_sizes, int n_in,
                              void* d_out, int out_size, void* d_ws, size_t ws_size,
                              hipStream_t stream) {
    const float* s            = (const float*)d_in[0];
    const float* y            = (const float*)d_in[1];
    const int*   ks           = (const int*)d_in[2];
    const int*   ts           = (const int*)d_in[3];
    const float* means        = (const float*)d_in[4];
    const float* covs         = (const float*)d_in[5];
    const float* b_mu         = (const float*)d_in[6];
    const float* b_log_sig    = (const float*)d_in[7];
    const float* beta_mu      = (const float*)d_in[8];
    const float* beta_log_sig = (const float*)d_in[9];

    float* ws   = (float*)d_ws;
    float* W    = ws + WS_W_OFF;
    float* lp   = ws + WS_LP_OFF;
    float* part = ws + WS_PART_OFF;

    k_build_w<<<1, 32, 0, stream>>>(means, covs, W);
    k_logpis<<<(N_K * N_T) / 256, 256, 0, stream>>>(y, b_mu, beta_mu, lp);
    k_prior<<<1, 256, 0, stream>>>(b_mu, b_log_sig, beta_mu, beta_log_sig, part);
    k_main<<<MAIN_BLOCKS, WAVES_PER_BLOCK * 32, 0, stream>>>(s, ks, ts, W, lp, part);
    k_final<<<1, 256, 0, stream>>>(part, (float*)d_out);
}